// LocalAveragePoolingSegmenter_72499047957271
// MI455X (gfx1250) — compile-verified
//
#include <hip/hip_runtime.h>
#include <stdint.h>

// Problem constants (from reference): B=16, T_AUDIO=4096, T_TEXT=1024, D=512
#define B_     16
#define TA     4096
#define TT     1024
#define DD     512
#define CHUNK  256
#define NC     (TA / CHUNK)   // 16 chunks along the audio axis
#define CSH    8              // log2(CHUNK)

typedef float vfloat4 __attribute__((ext_vector_type(4)));

// ---------------------------------------------------------------------------
// Pass 1: per-chunk inclusive prefix sums along the audio (frame) axis.
// Grid: B * NC = 256 blocks, 128 threads (4 wave32s). Each thread owns 4
// consecutive feature dims (float4) and scans 256 frames serially, so every
// global access is a b128 (512 B per wave per instruction, fully coalesced).
// Audio is loaded non-temporally: it is dead after this pass, and we want the
// 134 MB prefix array P to stay resident in the 192 MB L2 for pass 3.
// ---------------------------------------------------------------------------
__global__ __launch_bounds__(128) void lap_prefix_chunk(const float* __restrict__ audio,
                                                        float* __restrict__ P,
                                                        float* __restrict__ T) {
    const int bid = blockIdx.x;               // B*NC
    const int c   = bid & (NC - 1);
    const int b   = bid >> 4;                 // / NC
    const int d4  = threadIdx.x * 4;

    const size_t rowBase = ((size_t)b * TA + (size_t)c * CHUNK) * DD + d4;
    const float* __restrict__ src = audio + rowBase;
    float* __restrict__ dst = P + rowBase;

    vfloat4 s = {0.f, 0.f, 0.f, 0.f};
#pragma unroll 8
    for (int a = 0; a < CHUNK; ++a) {
        const vfloat4 x = __builtin_nontemporal_load((const vfloat4*)(src + (size_t)a * DD));
        s += x;
        *(vfloat4*)(dst + (size_t)a * DD) = s;
    }
    *(vfloat4*)(T + ((size_t)b * NC + c) * DD + d4) = s;
}

// ---------------------------------------------------------------------------
// Pass 2: exclusive scan of the 16 chunk totals -> O[b,c,d] (512 KB, L2-hot).
// Also copies text_token_len (2nd tuple output) into the output tail as raw
// int32 bits. Grid: B*D/4 threads = 8 blocks of 256.
// ---------------------------------------------------------------------------
__global__ __launch_bounds__(256) void lap_chunk_offsets(const float* __restrict__ T,
                                                         float* __restrict__ O,
                                                         const int* __restrict__ tok_len,
                                                         int* __restrict__ out_len) {
    const int gid = blockIdx.x * 256 + threadIdx.x;   // 0 .. B*D/4-1
    const int b  = gid >> 7;                          // DD/4 = 128 lanes per batch
    const int d4 = (gid & 127) * 4;

    vfloat4 run = {0.f, 0.f, 0.f, 0.f};
#pragma unroll
    for (int c = 0; c < NC; ++c) {
        const size_t i = ((size_t)b * NC + c) * DD + d4;
        *(vfloat4*)(O + i) = run;
        run += *(const vfloat4*)(T + i);
    }
    if (gid < B_) out_len[gid] = tok_len[gid];
}

// ---------------------------------------------------------------------------
// Pass 3: one wave32 per (b, token). Data-dependent row gather of the two
// prefix rows via CDNA5 async global->LDS copies (ASYNCcnt), then
// out[d] = (P[e,d] + O[ce,d] - P[s-1,d] - O[cs,d]) * 1/(e-s+1).
// LDS layout: lds[0..511] = prefix row e, lds[512..1023] = prefix row s-1.
// The ISA adds INST_OFFSET to BOTH the LDS and global address, so a single
// voffset VGPR + 4 literal offsets covers each 2 KB row.
// Output stores are non-temporal (never re-read; keep P in L2).
// ---------------------------------------------------------------------------
__global__ __launch_bounds__(32) void lap_gather(const float* __restrict__ P,
                                                 const float* __restrict__ O,
                                                 const int* __restrict__ tok_len,
                                                 const int* __restrict__ align2,
                                                 float* __restrict__ out) {
    __shared__ __align__(16) float lds[2 * DD];

    const int b    = blockIdx.x >> 10;        // / TT
    const int t    = blockIdx.x & (TT - 1);
    const int lane = threadIdx.x;

    float* __restrict__ orow = out + ((size_t)b * TT + t) * DD;
    const int len = tok_len[b];

    if (t >= len) {
        // Invalid token: reference yields zeros; d_out is poisoned, so write.
        const vfloat4 z = {0.f, 0.f, 0.f, 0.f};
#pragma unroll
        for (int k = 0; k < 4; ++k)
            __builtin_nontemporal_store(z, (vfloat4*)(orow + k * 128 + lane * 4));
        return;
    }

    const size_t ti = ((size_t)b * TT + t) * 2;
    const int s = align2[ti + 0];
    const int e = align2[ti + 1];
    const float inv = 1.0f / (float)(e - s + 1);
    const bool haveS = (s > 0);               // uniform across the wave

    const uint32_t ldsBase = (uint32_t)(uintptr_t)(void*)&lds[0]; // low 32b of flat = LDS offset
    const uint32_t vo   = (uint32_t)lane * 16u;                   // B128 per lane
    const uint32_t ldsE = ldsBase + vo;

    // Async copy prefix row e (2 KB = 4 x 512B wave-transfers) into the first half of lds
    const float* rowE = P + ((size_t)b * TA + e) * DD;
    asm volatile(
        "global_load_async_to_lds_b128 %0, %1, %2 offset:0\n\t"
        "global_load_async_to_lds_b128 %0, %1, %2 offset:512\n\t"
        "global_load_async_to_lds_b128 %0, %1, %2 offset:1024\n\t"
        "global_load_async_to_lds_b128 %0, %1, %2 offset:1536"
        :
        : "v"(ldsE), "v"(vo), "s"(rowE)
        : "memory");

    if (haveS) {
        const float* rowS = P + ((size_t)b * TA + (s - 1)) * DD;
        const uint32_t ldsS = ldsBase + 2048u + vo;
        asm volatile(
            "global_load_async_to_lds_b128 %0, %1, %2 offset:0\n\t"
            "global_load_async_to_lds_b128 %0, %1, %2 offset:512\n\t"
            "global_load_async_to_lds_b128 %0, %1, %2 offset:1024\n\t"
            "global_load_async_to_lds_b128 %0, %1, %2 offset:1536"
            :
            : "v"(ldsS), "v"(vo), "s"(rowS)
            : "memory");
    }

    // Chunk-offset rows (512 KB array, L2-resident): regular vector loads.
    const int ce = e >> CSH;
    const int cs = haveS ? ((s - 1) >> CSH) : 0;
    const float* __restrict__ OE = O + ((size_t)b * NC + ce) * DD;
    const float* __restrict__ OS = O + ((size_t)b * NC + cs) * DD;

    asm volatile("s_wait_asynccnt 0" ::: "memory");

#pragma unroll
    for (int k = 0; k < 4; ++k) {
        const int d = k * 128 + lane * 4;     // 16B-aligned, 512B/wave coalesced
        const vfloat4 pe = *(const vfloat4*)&lds[d];
        const vfloat4 oe = *(const vfloat4*)(OE + d);
        vfloat4 r;
        if (haveS) {
            const vfloat4 ps = *(const vfloat4*)&lds[DD + d];
            const vfloat4 os = *(const vfloat4*)(OS + d);
            r = (pe + oe - ps - os) * inv;
        } else {
            r = (pe + oe) * inv;
        }
        __builtin_nontemporal_store(r, (vfloat4*)(orow + d));
    }
}

// ---------------------------------------------------------------------------
// Workspace layout (bytes):
//   P : B*TA*D floats  = 134,217,728   (inclusive per-chunk prefix sums)
//   T : B*NC*D floats  =     524,288   (chunk totals)
//   O : B*NC*D floats  =     524,288   (exclusive chunk offsets)
// Total ~135.3 MB.
// ---------------------------------------------------------------------------
extern "C" void kernel_launch(void* const* d_in, const int* in_sizes, int n_in,
                              void* d_out, int out_size, void* d_ws, size_t ws_size,
                              hipStream_t stream) {
    (void)in_sizes; (void)n_in; (void)out_size; (void)ws_size;

    const float* audio   = (const float*)d_in[0];  // [B,TA,D] f32
    // d_in[1] audio_feats_len, d_in[2] text_token_for_audio,
    // d_in[3] text_token_embed_for_audio : unused by the reference math
    const int*   tok_len = (const int*)d_in[4];    // [B] i32
    const int*   align2  = (const int*)d_in[5];    // [B,TT,2] i32

    float* P = (float*)d_ws;
    float* T = P + (size_t)B_ * TA * DD;
    float* O = T + (size_t)B_ * NC * DD;

    float* out     = (float*)d_out;                        // [B,TT,D] f32
    int*   out_len = (int*)(out + (size_t)B_ * TT * DD);   // [B] i32 (tuple elem 2)

    lap_prefix_chunk <<<B_ * NC,          128, 0, stream>>>(audio, P, T);
    lap_chunk_offsets<<<(B_ * DD / 4)/256, 256, 0, stream>>>(T, O, tok_len, out_len);
    lap_gather       <<<B_ * TT,            32, 0, stream>>>(P, O, tok_len, align2, out);
}